// CredLightGCN_23854248362837
// MI455X (gfx1250) — compile-verified
//
#include <hip/hip_runtime.h>
#include <hip/hip_bf16.h>

typedef float v2f __attribute__((ext_vector_type(2)));
typedef float v8f __attribute__((ext_vector_type(8)));

#define EMB 64

// ---------------------------------------------------------------------------
// Zero a float4 array
__global__ void lgcn_zero4(float4* __restrict__ p, int n4) {
    int i = blockIdx.x * blockDim.x + threadIdx.x;
    if (i < n4) p[i] = make_float4(0.f, 0.f, 0.f, 0.f);
}

// Copy src into two destinations (init cur and sum with the base embeddings)
__global__ void lgcn_copy2(const float4* __restrict__ src,
                           float4* __restrict__ d1, float4* __restrict__ d2, int n4) {
    int i = blockIdx.x * blockDim.x + threadIdx.x;
    if (i < n4) {
        float4 v = src[i];
        d1[i] = v;
        d2[i] = v;
    }
}

// acc += add
__global__ void lgcn_accum4(float4* __restrict__ acc, const float4* __restrict__ add, int n4) {
    int i = blockIdx.x * blockDim.x + threadIdx.x;
    if (i < n4) {
        float4 a = acc[i], b = add[i];
        a.x += b.x; a.y += b.y; a.z += b.z; a.w += b.w;
        acc[i] = a;
    }
}

// ---------------------------------------------------------------------------
// SpMM scatter: dst[dst_idx[e]] += vals[e] * src[src_idx[e]]
// 16 lanes cooperate per edge, each lane moves 4 contiguous dims (b128 gather,
// 4x global_atomic_add_f32 scatter). All traffic is L2-resident (<192MB set).
__global__ void lgcn_spmm_scatter(const float* __restrict__ src,
                                  float* __restrict__ dst,
                                  const float* __restrict__ vals,
                                  const int* __restrict__ src_idx,
                                  const int* __restrict__ dst_idx,
                                  int n_edges) {
    int t = blockIdx.x * blockDim.x + threadIdx.x;
    int edge = t >> 4;        // 16 lanes per edge
    int sub  = t & 15;        // dims [sub*4, sub*4+4)
    if (edge >= n_edges) return;

    float v = vals[edge];
    size_t s = (size_t)src_idx[edge] * EMB + sub * 4;
    size_t d = (size_t)dst_idx[edge] * EMB + sub * 4;

    float4 a = *reinterpret_cast<const float4*>(src + s);  // global_load_b128
    float* dp = dst + d;
    atomicAdd(dp + 0, v * a.x);
    atomicAdd(dp + 1, v * a.y);
    atomicAdd(dp + 2, v * a.z);
    atomicAdd(dp + 3, v * a.w);
}

// ---------------------------------------------------------------------------
// Final scoring: out[b] = (1/16) * dot(sum_u[users[b]], sum_i[items[b]])
// One wave handles 16 pairs via chained V_WMMA_F32_16X16X4_F32:
//   C = A * B with A[m,k] = eu_m[k], B[k,n] = ei_n[k]  =>  C[m,m] = dot_m.
// A 16x4 f32 layout: lane l, VGPR j -> A[M=l%16, K=2*(l/16)+j]  (ISA 7.12.2)
// B 4x16 f32 layout (mirror): lane l, VGPR j -> B[K=2*(l/16)+j, N=l%16]
// So each lane loads the SAME (row, k0) float2 from eu for A and ei for B.
// C/D 16x16 f32: VGPR r, lanes 0-15 -> (M=r, N=lane); lanes 16-31 -> (M=r+8).
// Diagonal lives in lanes 0-7 (r=lane) and lanes 24-31 (r=lane-24).
__global__ void __launch_bounds__(32)
lgcn_dot_wmma(const float* __restrict__ sum_u, const float* __restrict__ sum_i,
              const int* __restrict__ users, const int* __restrict__ items,
              float* __restrict__ out, int batch) {
    const int group = blockIdx.x;          // 16 pairs per wave
    const int lane  = threadIdx.x;         // 0..31 (wave32)
    const int row   = lane & 15;           // pair within group
    const int half  = lane >> 4;           // K sub-pair selector

    int pair = group * 16 + row;
    int pc   = pair < batch ? pair : batch - 1;
    const float* urow = sum_u + (size_t)users[pc] * EMB;
    const float* irow = sum_i + (size_t)items[pc] * EMB;

    v8f c = {};
#pragma unroll
    for (int kb = 0; kb < EMB; kb += 4) {
        int k0 = kb + 2 * half;
        v2f a, b;
        a.x = urow[k0]; a.y = urow[k0 + 1];
        b.x = irow[k0]; b.y = irow[k0 + 1];
        // D = A x B + C  (16x16x4 f32)
        c = __builtin_amdgcn_wmma_f32_16x16x4_f32(
                /*neg_a=*/false, a, /*neg_b=*/false, b,
                /*c_mod=*/(short)0, c, /*reuse_a=*/false, /*reuse_b=*/false);
    }

    float cv[8];
#pragma unroll
    for (int r = 0; r < 8; ++r) cv[r] = c[r];

    const float scale = 1.0f / 16.0f;      // (1/(L+1))^2 with L=3
    if (lane < 8) {
        int p = group * 16 + lane;         // N = M = lane, VGPR r = lane
        if (p < batch) out[p] = cv[lane] * scale;
    } else if (lane >= 24) {
        int p = group * 16 + (lane - 16);  // N = M = lane-16, VGPR r = lane-24
        if (p < batch) out[p] = cv[lane - 24] * scale;
    }
}

// ---------------------------------------------------------------------------
extern "C" void kernel_launch(void* const* d_in, const int* in_sizes, int n_in,
                              void* d_out, int out_size, void* d_ws, size_t ws_size,
                              hipStream_t stream) {
    const float* user_emb  = (const float*)d_in[0];
    const float* item_emb  = (const float*)d_in[1];
    const float* edge_vals = (const float*)d_in[2];
    const int*   edge_u    = (const int*)d_in[3];
    const int*   edge_i    = (const int*)d_in[4];
    const int*   users     = (const int*)d_in[5];
    const int*   items     = (const int*)d_in[6];
    float*       out       = (float*)d_out;

    const int n_user_f = in_sizes[0];   // NUM_USERS * 64 floats
    const int n_item_f = in_sizes[1];   // NUM_ITEMS * 64 floats
    const int n_edges  = in_sizes[2];
    const int batch    = in_sizes[5];

    // Workspace layout: cur_u | nxt_u | sum_u | cur_i | nxt_i | sum_i
    float* ws    = (float*)d_ws;
    float* cur_u = ws;
    float* nxt_u = ws + (size_t)n_user_f;
    float* sum_u = ws + (size_t)2 * n_user_f;
    float* cur_i = ws + (size_t)3 * n_user_f;
    float* nxt_i = ws + (size_t)3 * n_user_f + (size_t)n_item_f;
    float* sum_i = ws + (size_t)3 * n_user_f + (size_t)2 * n_item_f;

    const int TB = 256;
    const int u4 = n_user_f / 4, i4 = n_item_f / 4;
    const int ub = (u4 + TB - 1) / TB, ib = (i4 + TB - 1) / TB;

    // Init: cur = sum = base embeddings
    lgcn_copy2<<<ub, TB, 0, stream>>>((const float4*)user_emb, (float4*)cur_u, (float4*)sum_u, u4);
    lgcn_copy2<<<ib, TB, 0, stream>>>((const float4*)item_emb, (float4*)cur_i, (float4*)sum_i, i4);

    const int spmm_blocks = (n_edges * 16 + TB - 1) / TB;

    for (int layer = 0; layer < 3; ++layer) {
        lgcn_zero4<<<ub, TB, 0, stream>>>((float4*)nxt_u, u4);
        lgcn_zero4<<<ib, TB, 0, stream>>>((float4*)nxt_i, i4);
        // new_i = M_ui @ e_u : gather user rows, scatter-add to items
        lgcn_spmm_scatter<<<spmm_blocks, TB, 0, stream>>>(cur_u, nxt_i, edge_vals,
                                                          edge_u, edge_i, n_edges);
        // new_u = M_iu @ e_i : gather item rows (OLD e_i), scatter-add to users
        lgcn_spmm_scatter<<<spmm_blocks, TB, 0, stream>>>(cur_i, nxt_u, edge_vals,
                                                          edge_i, edge_u, n_edges);
        lgcn_accum4<<<ub, TB, 0, stream>>>((float4*)sum_u, (const float4*)nxt_u, u4);
        lgcn_accum4<<<ib, TB, 0, stream>>>((float4*)sum_i, (const float4*)nxt_i, i4);
        // ping-pong (host-side, fully deterministic across calls)
        float* t;
        t = cur_u; cur_u = nxt_u; nxt_u = t;
        t = cur_i; cur_i = nxt_i; nxt_i = t;
    }

    // Scoring via f32 WMMA (16 pairs per wave)
    const int groups = (batch + 15) / 16;
    lgcn_dot_wmma<<<groups, 32, 0, stream>>>(sum_u, sum_i, users, items, out, batch);
}